// Sampler_58488864637011
// MI455X (gfx1250) — compile-verified
//
#include <hip/hip_runtime.h>
#include <math.h>

#define NT 1024            // threads per block (32 wave32 waves)
#define NB 2048            // histogram bins (11 bits)
#define PER (NB / NT)      // bins per thread in scans = 2
#define NW (NT / 32)       // 32 waves

#if defined(__HIP_DEVICE_COMPILE__) && __has_builtin(__builtin_amdgcn_global_load_async_to_lds_b128)
#define HAVE_ASYNC_LDS 1
#else
#define HAVE_ASYNC_LDS 0
#endif

#if HAVE_ASYNC_LDS
// Builtin signature (decoded from hipcc diagnostics):
//   param0 = AS1 v4i*  (global source, 128-bit vector)
//   param1 = AS3 v4i*  (LDS destination, 128-bit vector)
//   param2 = imm i32 offset, param3 = imm i32 cpol
typedef int v4i __attribute__((vector_size(16)));
typedef __attribute__((address_space(1))) v4i* async_gptr;
typedef __attribute__((address_space(3))) v4i* async_lptr;

template <int N>
__device__ __forceinline__ void wait_async_le() {
#if __has_builtin(__builtin_amdgcn_s_wait_asynccnt)
  __builtin_amdgcn_s_wait_asynccnt(N);
#else
  asm volatile("s_wait_asynccnt %0" ::"i"(N) : "memory");
#endif
}
#endif

// Order-preserving key, DESCENDING: larger float -> smaller key.
__device__ __forceinline__ unsigned int key_desc(float x) {
  unsigned int u = __float_as_uint(x);
  unsigned int m = (u & 0x80000000u) ? 0xFFFFFFFFu : 0x80000000u;
  return ~(u ^ m);
}

__device__ __forceinline__ float blockMaxF(float v, float* s_red) {
  const int lane = threadIdx.x & 31, wid = threadIdx.x >> 5;
#pragma unroll
  for (int off = 16; off > 0; off >>= 1) v = fmaxf(v, __shfl_xor(v, off, 32));
  if (lane == 0) s_red[wid] = v;
  __syncthreads();
  if (wid == 0) {
    float w = s_red[lane];
#pragma unroll
    for (int off = 16; off > 0; off >>= 1) w = fmaxf(w, __shfl_xor(w, off, 32));
    if (lane == 0) s_red[0] = w;
  }
  __syncthreads();
  float r = s_red[0];
  __syncthreads();
  return r;
}

__device__ __forceinline__ float blockSumF(float v, float* s_red) {
  const int lane = threadIdx.x & 31, wid = threadIdx.x >> 5;
#pragma unroll
  for (int off = 16; off > 0; off >>= 1) v += __shfl_xor(v, off, 32);
  if (lane == 0) s_red[wid] = v;
  __syncthreads();
  if (wid == 0) {
    float w = s_red[lane];
#pragma unroll
    for (int off = 16; off > 0; off >>= 1) w += __shfl_xor(w, off, 32);
    if (lane == 0) s_red[0] = w;
  }
  __syncthreads();
  float r = s_red[0];
  __syncthreads();
  return r;
}

__device__ __forceinline__ float blockRangeSumF(const float* a, int n, float* s_red) {
  float v = 0.0f;
  for (int j = threadIdx.x; j < n; j += NT) v += a[j];
  return blockSumF(v, s_red);
}

// Find smallest bin b with inclusive prefix sum(arr[0..b]) >= target.
// Results in *s_bin (-1 if total < target) and *s_before (exclusive prefix).
__device__ void scanFindU(const unsigned int* __restrict__ arr, unsigned int target,
                          unsigned int* s_red, int* s_bin, unsigned int* s_before) {
  const int tid = threadIdx.x, lane = tid & 31, wid = tid >> 5;
  unsigned int a[PER];
  unsigned int local = 0;
#pragma unroll
  for (int j = 0; j < PER; ++j) { a[j] = arr[tid * PER + j]; local += a[j]; }
  unsigned int v = local;
#pragma unroll
  for (int off = 1; off < 32; off <<= 1) {
    unsigned int n = (unsigned int)__shfl_up((int)v, off, 32);
    if (lane >= off) v += n;
  }
  if (lane == 31) s_red[wid] = v;
  if (tid == 0) *s_bin = -1;
  __syncthreads();
  if (wid == 0) {
    unsigned int w = s_red[lane];
#pragma unroll
    for (int off = 1; off < 32; off <<= 1) {
      unsigned int n = (unsigned int)__shfl_up((int)w, off, 32);
      if (lane >= off) w += n;
    }
    s_red[lane] = w;
  }
  __syncthreads();
  unsigned int incl = v + (wid ? s_red[wid - 1] : 0u);
  unsigned int excl = incl - local;
  if (excl < target && incl >= target) {
    unsigned int c = excl;
#pragma unroll
    for (int j = 0; j < PER; ++j) {
      unsigned int c2 = c + a[j];
      if (c2 >= target) { *s_bin = tid * PER + j; *s_before = c; break; }
      c = c2;
    }
  }
  __syncthreads();
}

__device__ void scanFindF(const float* __restrict__ arr, float target,
                          float* s_red, int* s_bin, float* s_before) {
  const int tid = threadIdx.x, lane = tid & 31, wid = tid >> 5;
  float a[PER];
  float local = 0.0f;
#pragma unroll
  for (int j = 0; j < PER; ++j) { a[j] = arr[tid * PER + j]; local += a[j]; }
  float v = local;
#pragma unroll
  for (int off = 1; off < 32; off <<= 1) {
    float n = __shfl_up(v, off, 32);
    if (lane >= off) v += n;
  }
  if (lane == 31) s_red[wid] = v;
  if (tid == 0) *s_bin = -1;
  __syncthreads();
  if (wid == 0) {
    float w = s_red[lane];
#pragma unroll
    for (int off = 1; off < 32; off <<= 1) {
      float n = __shfl_up(w, off, 32);
      if (lane >= off) w += n;
    }
    s_red[lane] = w;
  }
  __syncthreads();
  float incl = v + (wid ? s_red[wid - 1] : 0.0f);
  float excl = incl - local;
  if (excl < target && incl >= target) {
    float c = excl;
#pragma unroll
    for (int j = 0; j < PER; ++j) {
      float c2 = c + a[j];
      if (c2 >= target) { *s_bin = tid * PER + j; *s_before = c; break; }
      c = c2;
    }
  }
  __syncthreads();
}

extern "C" __global__ void __launch_bounds__(NT, 1)
sampler_rowwise(const float* __restrict__ logits, const float* __restrict__ temps,
                const int* __restrict__ topks, const float* __restrict__ topps,
                const float* __restrict__ noise, float* __restrict__ out, int V) {
  __shared__ unsigned int s_cnt[NB];
  __shared__ float s_exp12[NB];
  __shared__ float s_exp2[NB];
  __shared__ float s_redf[NW];
  __shared__ unsigned int s_redu[NW];
  __shared__ int s_redi[NW];
  __shared__ int s_bin;
  __shared__ float s_beforeF;
  __shared__ unsigned int s_beforeU;
#if HAVE_ASYNC_LDS
  __shared__ float s_tile[4][NT * 4];  // 4-deep ring of 16KB async tiles
#endif

  const int tid = threadIdx.x;
  const int lane = tid & 31;
  const int wid = tid >> 5;
  const int row = blockIdx.x;
  const float* __restrict__ lg = logits + (size_t)row * (size_t)V;
  const float* __restrict__ nz = noise + (size_t)row * (size_t)V;
  const float invT = 1.0f / temps[row];
  const int kk = topks[row];
  const bool kActive = (kk > 0 && kk < V);
  const unsigned int kTarget = kActive ? (unsigned int)kk : 0u;
  const float pval = topps[row];

  for (int j = tid; j < NB; j += NT) s_cnt[j] = 0u;
  __syncthreads();

  // ---- Pass 1 (HBM-bound): row max + level-1 count histogram ----
  float mymax = -INFINITY;
  auto p1elem = [&](float f) {
    float x = f * invT;
    mymax = fmaxf(mymax, x);
    if (kActive) atomicAdd(&s_cnt[key_desc(x) >> 21], 1u);
  };
#if HAVE_ASYNC_LDS
  {
    const int TILE = NT * 4;  // floats per tile (16KB)
    const int nFull = V / TILE;
    const char* gbase = (const char*)lg;
    // Prologue: issue up to 3 tiles ahead (keeps ~48KB/WG in flight to HBM).
#pragma unroll
    for (int pt = 0; pt < 3; ++pt) {
      if (pt < nFull) {
        __builtin_amdgcn_global_load_async_to_lds_b128(
            (async_gptr)(gbase + (size_t)pt * TILE * 4 + (size_t)tid * 16),
            (async_lptr)(&s_tile[pt][tid * 4]), 0, 0);
      }
    }
#pragma unroll 1
    for (int t = 0; t < nFull; ++t) {
      if (t + 3 < nFull) {
        __builtin_amdgcn_global_load_async_to_lds_b128(
            (async_gptr)(gbase + (size_t)(t + 3) * TILE * 4 + (size_t)tid * 16),
            (async_lptr)(&s_tile[(t + 3) & 3][tid * 4]), 0, 0);
        wait_async_le<3>();
      } else {
        int rem = nFull - 1 - t;  // tiles still outstanding beyond t (0..2)
        if (rem >= 2) wait_async_le<2>();
        else if (rem == 1) wait_async_le<1>();
        else wait_async_le<0>();
      }
      __syncthreads();
      const float* buf = &s_tile[t & 3][tid * 4];
      p1elem(buf[0]); p1elem(buf[1]); p1elem(buf[2]); p1elem(buf[3]);
      __syncthreads();
    }
    for (int i = nFull * TILE + tid; i < V; i += NT) p1elem(lg[i]);
  }
#else
  {
    const int nv = V >> 2;
    const float4* lg4 = (const float4*)lg;
    for (int i = tid; i < nv; i += NT) {
      __builtin_prefetch(lg4 + i + NT * 4, 0, 1);
      float4 f = lg4[i];
      p1elem(f.x); p1elem(f.y); p1elem(f.z); p1elem(f.w);
    }
    for (int i = (nv << 2) + tid; i < V; i += NT) p1elem(lg[i]);
  }
#endif
  __syncthreads();

  const float M = blockMaxF(mymax, s_redf);
  const unsigned int dMin = key_desc(M);

  // ---- top-k level-1 boundary ----
  int bK = NB;
  unsigned int needK = 0u;
  if (kActive) {
    scanFindU(s_cnt, kTarget, s_redu, &s_bin, &s_beforeU);
    bK = s_bin;
    needK = kTarget - s_beforeU;
  }
  __syncthreads();
  for (int j = tid; j < NB; j += NT) { s_cnt[j] = 0u; s_exp12[j] = 0.0f; s_exp2[j] = 0.0f; }
  __syncthreads();

  // ---- Pass 2 (L2-hot): exp-sums per bin above boundary + refinement in bin bK ----
  auto p2elem = [&](float f) {
    float x = f * invT;
    unsigned int d = key_desc(x);
    int b = (int)(d >> 21);
    if (b < bK) {
      atomicAdd(&s_exp12[b], __expf(x - M));
    } else if (kActive && b == bK) {
      int sb = (int)((d >> 10) & (unsigned int)(NB - 1));
      atomicAdd(&s_cnt[sb], 1u);
      atomicAdd(&s_exp2[sb], __expf(x - M));
    }
  };
  {
    const int nv = V >> 2;
    const float4* lg4 = (const float4*)lg;
#pragma unroll 2
    for (int i = tid; i < nv; i += NT) {
      __builtin_prefetch(lg4 + i + NT * 2, 0, 1);
      float4 f = lg4[i];
      p2elem(f.x); p2elem(f.y); p2elem(f.z); p2elem(f.w);
    }
    for (int i = (nv << 2) + tid; i < V; i += NT) p2elem(lg[i]);
  }
  __syncthreads();

  unsigned int dK = 0xFFFFFFFFu;
  int b2k = -1;
  float expAbove, ZK;
  if (kActive) {
    scanFindU(s_cnt, needK, s_redu, &s_bin, &s_beforeU);
    b2k = s_bin;
    dK = (((unsigned int)bK) << 21) | (((unsigned int)b2k) << 10) | 0x3FFu;
    expAbove = blockRangeSumF(s_exp12, NB, s_redf);          // exp-sum of bins < bK
    float withinK = blockRangeSumF(s_exp2, b2k + 1, s_redf); // kept part of bin bK
    ZK = expAbove + withinK;                                 // softmax denom of top-k set
  } else {
    expAbove = blockRangeSumF(s_exp12, NB, s_redf);
    ZK = expAbove;
  }

  // ---- top-p threshold: smallest value with cum exp-sum < p*ZK ----
  const float P = pval * ZK;
  unsigned int dFin;
  if (!(P > 0.0f)) {
    dFin = 0u;  // keep only the max token (fixed up by dMin below)
  } else if (!(P < ZK)) {
    dFin = dK;  // p keeps the whole top-k set
  } else if (!kActive || expAbove >= P) {
    scanFindF(s_exp12, P, s_redf, &s_bin, &s_beforeF);
    int bP = s_bin;
    float cumBefore = s_beforeF;
    if (bP < 0) {
      dFin = dK;
    } else {
      __syncthreads();
      for (int j = tid; j < NB; j += NT) s_exp2[j] = 0.0f;
      __syncthreads();
      // Pass 3 (L2-hot, conditional): refine exp-sums inside the p-boundary bin
      auto p3elem = [&](float f) {
        float x = f * invT;
        unsigned int d = key_desc(x);
        if ((int)(d >> 21) == bP)
          atomicAdd(&s_exp2[(d >> 10) & (unsigned int)(NB - 1)], __expf(x - M));
      };
      const int nv = V >> 2;
      const float4* lg4 = (const float4*)lg;
#pragma unroll 2
      for (int i = tid; i < nv; i += NT) {
        __builtin_prefetch(lg4 + i + NT * 2, 0, 1);
        float4 f = lg4[i];
        p3elem(f.x); p3elem(f.y); p3elem(f.z); p3elem(f.w);
      }
      for (int i = (nv << 2) + tid; i < V; i += NT) p3elem(lg[i]);
      __syncthreads();
      scanFindF(s_exp2, P - cumBefore, s_redf, &s_bin, &s_beforeF);
      int b2p = s_bin;
      if (b2p < 0) {
        dFin = (((unsigned int)bP) << 21) | 0x1FFFFFu;  // keep whole bin (fp slack)
      } else {
        unsigned int cut = (((unsigned int)bP) << 21) | (((unsigned int)b2p) << 10);
        dFin = (cut == 0u) ? 0u : (cut - 1u);
      }
      if (dFin > dK) dFin = dK;
    }
  } else {
    // boundary lies inside bin bK: reuse its refinement exp-sums (s_exp2)
    scanFindF(s_exp2, P - expAbove, s_redf, &s_bin, &s_beforeF);
    int b2p = s_bin;
    if (b2p < 0 || b2p > b2k) {
      dFin = dK;
    } else {
      unsigned int cut = (((unsigned int)bK) << 21) | (((unsigned int)b2p) << 10);
      unsigned int c = (cut == 0u) ? 0u : (cut - 1u);
      dFin = (c < dK) ? c : dK;
    }
  }
  if (dFin < dMin) dFin = dMin;  // always keep the top token
  __syncthreads();

  // ---- Pass 4: Gumbel-max argmax over kept set; noise read only when kept ----
  float best = -INFINITY;
  int bestIdx = 0;
  auto p4elem = [&](float f, int idx) {
    float x = f * invT;
    unsigned int d = key_desc(x);
    if (d <= dFin) {
      float u = nz[idx];
      float q = fmaxf(-__logf(u), 1e-10f);
      float s = x - __logf(q);
      if (s > best || (s == best && idx < bestIdx)) { best = s; bestIdx = idx; }
    }
  };
  {
    const int nv = V >> 2;
    const float4* lg4 = (const float4*)lg;
#pragma unroll 2
    for (int i = tid; i < nv; i += NT) {
      __builtin_prefetch(lg4 + i + NT * 2, 0, 1);
      float4 f = lg4[i];
      int base = i << 2;
      p4elem(f.x, base); p4elem(f.y, base + 1); p4elem(f.z, base + 2); p4elem(f.w, base + 3);
    }
    for (int i = (nv << 2) + tid; i < V; i += NT) p4elem(lg[i], i);
  }
#pragma unroll
  for (int off = 16; off > 0; off >>= 1) {
    float s2 = __shfl_xor(best, off, 32);
    int i2 = __shfl_xor(bestIdx, off, 32);
    if (s2 > best || (s2 == best && i2 < bestIdx)) { best = s2; bestIdx = i2; }
  }
  if (lane == 0) { s_redf[wid] = best; s_redi[wid] = bestIdx; }
  __syncthreads();
  if (wid == 0) {
    float b = s_redf[lane];
    int bi2 = s_redi[lane];
#pragma unroll
    for (int off = 16; off > 0; off >>= 1) {
      float s2 = __shfl_xor(b, off, 32);
      int i2 = __shfl_xor(bi2, off, 32);
      if (s2 > b || (s2 == b && i2 < bi2)) { b = s2; bi2 = i2; }
    }
    if (lane == 0) out[row] = (float)bi2;
  }
}

extern "C" void kernel_launch(void* const* d_in, const int* in_sizes, int n_in,
                              void* d_out, int out_size, void* d_ws, size_t ws_size,
                              hipStream_t stream) {
  (void)n_in; (void)d_ws; (void)ws_size; (void)out_size;
  const float* logits = (const float*)d_in[0];
  const float* temps  = (const float*)d_in[1];
  const int*   topks  = (const int*)d_in[2];   // JAX default demotes int64 -> int32
  const float* topps  = (const float*)d_in[3];
  const float* noise  = (const float*)d_in[4];
  float* out = (float*)d_out;
  const int Bn = in_sizes[1];
  const int V = (Bn > 0) ? in_sizes[0] / Bn : 0;
  if (Bn <= 0 || V <= 0) return;
  sampler_rowwise<<<Bn, NT, 0, stream>>>(logits, temps, topks, topps, noise, out, V);
}